// LimbLength_80831284510869
// MI455X (gfx1250) — compile-verified
//
#include <hip/hip_runtime.h>
#include <math.h>

#define THREADS          256
#define ELEMS_PER_BLOCK  256
#define ROW_FLOATS       45                      // 15 joints * 3 coords
#define ROW_BYTES        180
#define BLOCK_BYTES      (ELEMS_PER_BLOCK * ROW_BYTES)       // 46080
#define FULL_PASSES      (BLOCK_BYTES / (THREADS * 16))      // 11
#define TAIL_LANES       ((BLOCK_BYTES - FULL_PASSES * THREADS * 16) / 16) // 64

__device__ __forceinline__ void limb_compute(const float* __restrict__ v, float* r) {
    constexpr int conn[15] = {0, 0, 1, 1, 1, 3, 4, 5, 6, 2, 2, 9, 10, 11, 12};
#pragma unroll
    for (int k = 0; k < 15; ++k) {
        const int c = conn[k];
        const float dx = v[3 * k + 0] - v[3 * c + 0];
        const float dy = v[3 * k + 1] - v[3 * c + 1];
        const float dz = v[3 * k + 2] - v[3 * c + 2];
        r[k] = sqrtf(dx * dx + dy * dy + dz * dz);
    }
}

__global__ __launch_bounds__(THREADS) void LimbLength_kernel(const float* __restrict__ in,
                                                             float* __restrict__ out,
                                                             int B) {
    __shared__ __align__(16) float smem[BLOCK_BYTES / 4];
    const int tid  = threadIdx.x;
    const int base = blockIdx.x * ELEMS_PER_BLOCK;

#if defined(__gfx1250__)
    if (base + ELEMS_PER_BLOCK <= B) {
        // ---- Async DMA stage: identity copy of this block's 46080B slab into LDS ----
        // Per-lane: LDS byte address in a VGPR, full 64-bit global address in a VGPR pair (GV mode).
        unsigned lds = (unsigned)(uintptr_t)(&smem[0]) + (unsigned)(tid * 16);
        unsigned long long g = (unsigned long long)(uintptr_t)in
                             + (unsigned long long)base * (unsigned long long)ROW_BYTES
                             + (unsigned long long)(tid * 16);
#pragma unroll
        for (int p = 0; p < FULL_PASSES; ++p) {
            asm volatile("global_load_async_to_lds_b128 %0, %1, off"
                         :: "v"(lds), "v"(g) : "memory");
            lds += THREADS * 16;
            g   += THREADS * 16;
        }
        if (tid < TAIL_LANES) {  // remaining 1024 bytes
            asm volatile("global_load_async_to_lds_b128 %0, %1, off"
                         :: "v"(lds), "v"(g) : "memory");
        }
        asm volatile("s_wait_asynccnt 0" ::: "memory");
        __syncthreads();

        // ---- Compute: thread t owns skeleton (base + t); 45 floats at LDS offset t*180.
        // (45*t + j) mod 64 is distinct across lanes (45 odd) -> conflict-free ds_load_b32.
        float v[ROW_FLOATS];
        const float* row = &smem[tid * ROW_FLOATS];
#pragma unroll
        for (int j = 0; j < ROW_FLOATS; ++j) v[j] = row[j];

        float r[15];
        limb_compute(v, r);

        float* o = out + (size_t)(base + tid) * 15;
#pragma unroll
        for (int k = 0; k < 15; ++k) o[k] = r[k];
        return;
    }
#endif
    // ---- Generic / partial-block fallback: direct global loads ----
    const int b = base + tid;
    if (b < B) {
        const float* p = in + (size_t)b * ROW_FLOATS;
        float v[ROW_FLOATS];
#pragma unroll
        for (int j = 0; j < ROW_FLOATS; ++j) v[j] = p[j];

        float r[15];
        limb_compute(v, r);

        float* o = out + (size_t)b * 15;
#pragma unroll
        for (int k = 0; k < 15; ++k) o[k] = r[k];
    }
}

extern "C" void kernel_launch(void* const* d_in, const int* in_sizes, int n_in,
                              void* d_out, int out_size, void* d_ws, size_t ws_size,
                              hipStream_t stream) {
    const float* in  = (const float*)d_in[0];
    float*       out = (float*)d_out;
    const int B      = in_sizes[0] / ROW_FLOATS;          // 1,000,000
    const int blocks = (B + ELEMS_PER_BLOCK - 1) / ELEMS_PER_BLOCK;
    LimbLength_kernel<<<blocks, THREADS, 0, stream>>>(in, out, B);
}